// GCN_A_64493228917217
// MI455X (gfx1250) — compile-verified
//
#include <hip/hip_runtime.h>
#include <math.h>

#define N_NODES 50000
#define N_EDGES 800000
#define F_IN    512
#define F_HID   256
#define F_OUT   64
#define LN_EPS  1e-5f
#define L2_EPS  1e-12f

typedef __attribute__((ext_vector_type(2))) float v2f;
typedef __attribute__((ext_vector_type(8))) float v8f;

__device__ __forceinline__ float wave_sum32(float v) {
#pragma unroll
    for (int m = 16; m >= 1; m >>= 1) v += __shfl_xor(v, m, 32);
    return v;
}

// ---------------------------------------------------------------- degree init
__global__ void __launch_bounds__(256) k_deg_init(float* deg) {
    int i = blockIdx.x * 256 + threadIdx.x;
    if (i < N_NODES) deg[i] = 1.0f;           // self loop contributes 1
}

__global__ void __launch_bounds__(256) k_deg_accum(const int* __restrict__ dst,
                                                   float* deg) {
    int e = blockIdx.x * 256 + threadIdx.x;
    if (e < N_EDGES) atomicAdd(&deg[dst[e]], 1.0f);
}

__global__ void __launch_bounds__(256) k_dinv(float* deg) {
    int i = blockIdx.x * 256 + threadIdx.x;
    if (i < N_NODES) deg[i] = rsqrtf(deg[i]); // deg >= 1 always (self loops)
}

// ------------------------------------------------- nan_to_num + row L2 norm
// one wave per node; 512 floats = 4 float4 per lane
__global__ void __launch_bounds__(256) k_l2norm(const float* __restrict__ x,
                                                float* __restrict__ xn) {
    const int node = blockIdx.x * 8 + (threadIdx.x >> 5);
    const int lane = threadIdx.x & 31;
    const float4* row = (const float4*)(x + (size_t)node * F_IN);
    float4 v[4];
    float ss = 0.0f;
#pragma unroll
    for (int i = 0; i < 4; ++i) {
        float4 t = row[lane + 32 * i];
        t.x = isfinite(t.x) ? t.x : 0.0f;
        t.y = isfinite(t.y) ? t.y : 0.0f;
        t.z = isfinite(t.z) ? t.z : 0.0f;
        t.w = isfinite(t.w) ? t.w : 0.0f;
        v[i] = t;
        ss += t.x * t.x + t.y * t.y + t.z * t.z + t.w * t.w;
    }
    ss = wave_sum32(ss);
    float scale = 1.0f / fmaxf(sqrtf(ss), L2_EPS);
    float4* orow = (float4*)(xn + (size_t)node * F_IN);
#pragma unroll
    for (int i = 0; i < 4; ++i) {
        float4 t = v[i];
        t.x *= scale; t.y *= scale; t.z *= scale; t.w *= scale;
        orow[lane + 32 * i] = t;
    }
}

// ---------------------------------------------------------- fp32 WMMA GEMM
// C[M x NOUT] = A[M x K] * B[K x NOUT], all row-major, M multiple of 16,
// NOUT multiple of 64. 128 threads = 4 waves; block covers 16 rows x 64 cols.
// A tile staged in LDS; uses V_WMMA_F32_16X16X4_F32 (K-step 4).
template <int K, int NOUT>
__global__ void __launch_bounds__(128) k_gemm_wmma(const float* __restrict__ A,
                                                   const float* __restrict__ B,
                                                   float* __restrict__ C) {
    __shared__ float As[16 * K];
    const int tid  = threadIdx.x;
    const int m0   = blockIdx.x * 16;
    const int wave = tid >> 5;
    const int lane = tid & 31;
    const int n0   = blockIdx.y * 64 + wave * 16;

    // cooperative coalesced stage of the 16xK A tile (float4 granularity)
    {
        const float4* gsrc = (const float4*)A + (size_t)m0 * (K / 4);
        float4* lds = (float4*)As;
#pragma unroll 4
        for (int i = tid; i < 16 * (K / 4); i += 128) lds[i] = gsrc[i];
    }
    __syncthreads();

    const int r   = lane & 15;          // A row within tile / C column
    const int kk  = (lane >> 4) * 2;    // K sub-offset: 0 for lanes 0-15, 2 for 16-31

    v8f acc = {};
#pragma unroll 4
    for (int k = 0; k < K; k += 4) {
        v2f a, b;
        a.x = As[r * K + k + kk];
        a.y = As[r * K + k + kk + 1];
        b.x = B[(size_t)(k + kk) * NOUT + n0 + r];
        b.y = B[(size_t)(k + kk + 1) * NOUT + n0 + r];
        acc = __builtin_amdgcn_wmma_f32_16x16x4_f32(
            /*neg_a=*/false, a, /*neg_b=*/false, b,
            /*c_mod=*/(short)0, acc, /*reuse_a=*/false, /*reuse_b=*/false);
    }

    // C/D layout: lanes 0-15 -> rows 0..7, lanes 16-31 -> rows 8..15, col = lane&15
    const int rb = (lane >> 4) * 8;
#pragma unroll
    for (int i = 0; i < 8; ++i)
        C[(size_t)(m0 + rb + i) * NOUT + n0 + r] = acc[i];
}

// ------------------------------------------------ edge scatter-add (gather+atomic)
// one wave per edge (incl. N self loops appended logically at the end)
template <int F>
__global__ void __launch_bounds__(256) k_scatter(const float* __restrict__ H,
                                                 const int* __restrict__ srcp,
                                                 const int* __restrict__ dstp,
                                                 const float* __restrict__ dinv,
                                                 float* __restrict__ out) {
    const int wave = blockIdx.x * 8 + (threadIdx.x >> 5);
    const int lane = threadIdx.x & 31;
    const int Etot = N_EDGES + N_NODES;
    if (wave >= Etot) return;
    int s, d; float w;
    if (wave < N_EDGES) {
        s = srcp[wave]; d = dstp[wave];
        w = dinv[s] * dinv[d];
    } else {
        s = d = wave - N_EDGES;
        float di = dinv[s];
        w = di * di;
    }
    const float4* hr = (const float4*)(H + (size_t)s * F);
    float* orow = out + (size_t)d * F;
    __builtin_prefetch(hr, 0, 0);
#pragma unroll
    for (int j = lane; j < F / 4; j += 32) {
        float4 v = hr[j];
        atomicAdd(&orow[4 * j + 0], v.x * w);
        atomicAdd(&orow[4 * j + 1], v.y * w);
        atomicAdd(&orow[4 * j + 2], v.z * w);
        atomicAdd(&orow[4 * j + 3], v.w * w);
    }
}

// ---------------------------------------------- bias + LayerNorm + ReLU in place
// one wave per node over F_HID=256 values (2 float4 per lane)
__global__ void __launch_bounds__(256) k_ln_relu(float* __restrict__ H,
                                                 const float* __restrict__ b1,
                                                 const float* __restrict__ gamma,
                                                 const float* __restrict__ beta) {
    const int node = blockIdx.x * 8 + (threadIdx.x >> 5);
    const int lane = threadIdx.x & 31;
    float* row = H + (size_t)node * F_HID;
    float4 v[2];
    float s = 0.0f;
#pragma unroll
    for (int i = 0; i < 2; ++i) {
        float4 t  = ((float4*)row)[lane + 32 * i];
        float4 bb = ((const float4*)b1)[lane + 32 * i];
        t.x += bb.x; t.y += bb.y; t.z += bb.z; t.w += bb.w;
        v[i] = t;
        s += t.x + t.y + t.z + t.w;
    }
    float mu = wave_sum32(s) * (1.0f / F_HID);
    float ss = 0.0f;
#pragma unroll
    for (int i = 0; i < 2; ++i) {
        float4 t = v[i];
        ss += (t.x - mu) * (t.x - mu) + (t.y - mu) * (t.y - mu) +
              (t.z - mu) * (t.z - mu) + (t.w - mu) * (t.w - mu);
    }
    float rstd = rsqrtf(wave_sum32(ss) * (1.0f / F_HID) + LN_EPS);
#pragma unroll
    for (int i = 0; i < 2; ++i) {
        float4 t = v[i];
        float4 g = ((const float4*)gamma)[lane + 32 * i];
        float4 b = ((const float4*)beta)[lane + 32 * i];
        t.x = fmaxf((t.x - mu) * rstd * g.x + b.x, 0.0f);
        t.y = fmaxf((t.y - mu) * rstd * g.y + b.y, 0.0f);
        t.z = fmaxf((t.z - mu) * rstd * g.z + b.z, 0.0f);
        t.w = fmaxf((t.w - mu) * rstd * g.w + b.w, 0.0f);
        ((float4*)row)[lane + 32 * i] = t;
    }
}

// ----------------------------------------------------- final bias + nan_to_num
__global__ void __launch_bounds__(256) k_finalize(float* __restrict__ out,
                                                  const float* __restrict__ b2) {
    int i = blockIdx.x * 256 + threadIdx.x;
    if (i >= N_NODES * F_OUT) return;
    float v = out[i] + b2[i & (F_OUT - 1)];
    if (isnan(v)) v = 0.0f;
    else if (isinf(v)) v = v > 0.0f ? 20.0f : -20.0f;
    out[i] = v;
}

extern "C" void kernel_launch(void* const* d_in, const int* in_sizes, int n_in,
                              void* d_out, int out_size, void* d_ws, size_t ws_size,
                              hipStream_t stream) {
    const float* x     = (const float*)d_in[0];
    const int*   ei    = (const int*)d_in[1];
    const float* W1    = (const float*)d_in[2];
    const float* b1    = (const float*)d_in[3];
    const float* gamma = (const float*)d_in[4];
    const float* beta  = (const float*)d_in[5];
    const float* W2    = (const float*)d_in[6];
    const float* b2    = (const float*)d_in[7];
    const float* Wskip = (const float*)d_in[8];
    const int* srcp = ei;
    const int* dstp = ei + N_EDGES;

    float* ws   = (float*)d_ws;
    float* dinv = ws;                                   // N
    float* xn   = dinv + N_NODES;                       // N * 512
    float* h1   = xn + (size_t)N_NODES * F_IN;          // N * 256 (later reused for h2)
    float* agg  = h1 + (size_t)N_NODES * F_HID;         // N * 256
    float* out  = (float*)d_out;                        // N * 64

    const int Etot = N_EDGES + N_NODES;

    // degree / norm
    hipMemsetAsync(agg, 0, (size_t)N_NODES * F_HID * sizeof(float), stream);
    k_deg_init<<<(N_NODES + 255) / 256, 256, 0, stream>>>(dinv);
    k_deg_accum<<<(N_EDGES + 255) / 256, 256, 0, stream>>>(dstp, dinv);
    k_dinv<<<(N_NODES + 255) / 256, 256, 0, stream>>>(dinv);

    // input clean + L2 normalize
    k_l2norm<<<N_NODES / 8, 256, 0, stream>>>(x, xn);

    // h1 = xn @ W1 ; skip = xn @ Wskip (straight into d_out)
    k_gemm_wmma<F_IN, F_HID><<<dim3(N_NODES / 16, F_HID / 64), 128, 0, stream>>>(xn, W1, h1);
    k_gemm_wmma<F_IN, F_OUT><<<dim3(N_NODES / 16, F_OUT / 64), 128, 0, stream>>>(xn, Wskip, out);

    // agg1 = scatter(h1) ; + b1 ; LN ; ReLU (in place)
    k_scatter<F_HID><<<(Etot + 7) / 8, 256, 0, stream>>>(h1, srcp, dstp, dinv, agg);
    k_ln_relu<<<N_NODES / 8, 256, 0, stream>>>(agg, b1, gamma, beta);

    // h2 = agg @ W2 (into h1 buffer) ; d_out += scatter(h2)
    k_gemm_wmma<F_HID, F_OUT><<<dim3(N_NODES / 16, F_OUT / 64), 128, 0, stream>>>(agg, W2, h1);
    k_scatter<F_OUT><<<(Etot + 7) / 8, 256, 0, stream>>>(h1, srcp, dstp, dinv, out);

    // + b2, nan_to_num(nan=0, +inf=20, -inf=-20)
    k_finalize<<<(N_NODES * F_OUT + 255) / 256, 256, 0, stream>>>(out, b2);
}